// EquilibriumPropagationNet_9552007266769
// MI455X (gfx1250) — compile-verified
//
#include <hip/hip_runtime.h>
#include <hip/hip_bf16.h>

typedef __attribute__((ext_vector_type(2))) float v2f;
typedef __attribute__((ext_vector_type(8))) float v8f;
typedef __attribute__((ext_vector_type(4))) unsigned int v4u;
typedef __attribute__((ext_vector_type(8))) int v8i;
typedef __attribute__((ext_vector_type(4))) int v4i;

#define D_IN  784
#define HID   1024
#define D_OUT 10
#define MT    16          // batch rows per workgroup
#define NTHREADS 128      // 4 wave32
#define W2P   16          // W2 stored [HID][16], zero-padded j >= 10

__device__ __forceinline__ float sigmoidf(float v) {
    return 1.0f / (1.0f + __expf(-v));
}

__global__ __launch_bounds__(NTHREADS)
void eqprop_settle_kernel(const float* __restrict__ xg,
                          const float* __restrict__ u1g,
                          const float* __restrict__ u2g,
                          const float* __restrict__ W1g,
                          const float* __restrict__ W2g,
                          const float* __restrict__ b1g,
                          const float* __restrict__ b2g,
                          const int*   __restrict__ steps_p,
                          float* __restrict__ out)
{
    extern __shared__ char smem[];
    float* xw1_lds  = (float*)smem;                  // MT*HID   (64 KB)
    float* u1_lds   = xw1_lds + MT * HID;            // MT*HID   (64 KB; x staging fits)
    float* r1_lds   = u1_lds  + MT * HID;            // MT*HID   (64 KB)
    float* w2_lds   = r1_lds  + MT * HID;            // HID*W2P  (64 KB, zero-padded)
    float* b1_lds   = w2_lds  + HID * W2P;           // HID
    float* b2_lds   = b1_lds  + HID;                 // 16
    float* u2_lds   = b2_lds  + 16;                  // MT*16
    float* r2_lds   = u2_lds  + MT * 16;             // MT*16 (zero-padded)
    float* sred_lds = r2_lds  + MT * 16;             // 4 waves * 256

    const int tid  = threadIdx.x;
    const int row0 = blockIdx.x * MT;
    const int wave  = tid >> 5;
    const int lane  = tid & 31;
    const int laneM = lane & 15;
    const int hi    = lane >> 4;          // fragment half: K offset 0 or 2

    // ---------------- stage x tile into the u1 LDS region (b128) ---------
    float* xstage = u1_lds;               // MT*D_IN floats (50 KB)
    {
        const float4* xg4 = (const float4*)(xg + (size_t)row0 * D_IN);
        float4* xs4 = (float4*)xstage;
        for (int i = tid; i < MT * D_IN / 4; i += NTHREADS)
            xs4[i] = xg4[i];
    }
    __syncthreads();

    // ---------------- Phase A: xW1 tile via V_WMMA_F32_16X16X4_F32 -------
    for (int nt = wave; nt < HID / 16; nt += 4) {
        const int n0 = nt * 16;
        v8f acc = {};
        const float* xp  = xstage + laneM * D_IN + hi * 2;
        const float* w1p = W1g + (size_t)(hi * 2) * HID + n0 + laneM;
        for (int k0 = 0; k0 < D_IN; k0 += 4) {
            v2f a; a.x = xp[0];  a.y = xp[1];
            v2f b; b.x = w1p[0]; b.y = w1p[HID];
            acc = __builtin_amdgcn_wmma_f32_16x16x4_f32(
                false, a, false, b, (short)0, acc, false, false);
            xp  += 4;
            w1p += (size_t)4 * HID;
            __builtin_prefetch(w1p, 0, 0);   // global_prefetch_b8: stream W1 (L2-resident)
        }
        #pragma unroll
        for (int i = 0; i < 8; ++i)
            xw1_lds[(i + hi * 8) * HID + n0 + laneM] = acc[i];
    }
    __syncthreads();

    // ---------------- TDM: DMA the u1 tile (64 KB contiguous) into LDS ---
    // 2D descriptor, 1 row of 16384 f32; runs concurrently with the W2/b1/u2
    // cooperative loads below; completion via TENSORcnt.
    if (wave == 0) {
        const unsigned long long gaddr =
            (unsigned long long)(const void*)(u1g + (size_t)row0 * HID);
        const unsigned int ldsoff =
            (unsigned int)((const char*)u1_lds - (const char*)smem);
        v4u g0;
        g0.x = 0x1u;                                      // count=1 (valid user D#)
        g0.y = ldsoff;                                    // lds_addr (bytes)
        g0.z = (unsigned int)(gaddr & 0xFFFFFFFFu);       // global_addr[31:0]
        g0.w = (unsigned int)(gaddr >> 32) | (2u << 30);  // global_addr[56:32] | type=2
        v8i g1;
        g1[0] = 0x00020000;                  // workgroup_mask=0, data_size=2 (4 bytes)
        g1[1] = (int)((MT * HID) << 16);     // tensor_dim0[15:0] = 16384
        g1[2] = (int)(1u << 16);             // tensor_dim0 hi = 0, tensor_dim1 = 1
        g1[3] = (int)((MT * HID) << 16);     // tile_dim0 = 16384
        g1[4] = 1;                           // tile_dim1 = 1, tile_dim2 = 0
        g1[5] = MT * HID;                    // tensor_dim0_stride = 16384
        g1[6] = 0;
        g1[7] = 0;
        const v4i gz4 = {0, 0, 0, 0};        // groups 2/3 unused (2D tensor)
        const v8i gz8 = {0, 0, 0, 0, 0, 0, 0, 0};
        __builtin_amdgcn_tensor_load_to_lds(g0, g1, gz4, gz4, gz8, 0);
        __builtin_amdgcn_s_wait_tensorcnt(0);
    }

    // ---------------- load weights / u2 into LDS (overlaps the TDM) ------
    for (int i = tid; i < HID * W2P; i += NTHREADS) {
        const int h = i >> 4, j = i & 15;
        w2_lds[i] = (j < D_OUT) ? W2g[h * D_OUT + j] : 0.0f;
    }
    for (int i = tid; i < HID; i += NTHREADS) b1_lds[i] = b1g[i];
    if (tid < D_OUT) b2_lds[tid] = b2g[tid];
    for (int i = tid; i < MT * 16; i += NTHREADS) {
        const int mm = i >> 4, j = i & 15;
        u2_lds[i] = (j < D_OUT) ? u2g[(size_t)row0 * D_OUT + mm * D_OUT + j] : 0.0f;
    }
    __syncthreads();

    // ---------------- Phase B: settle steps, GEMMs on the matrix pipe ----
    const int nsteps = steps_p[0];

    for (int st = 0; st < nsteps; ++st) {
        // r2 = sigmoid(u2), zero-padded (u2 stable here)
        for (int i = tid; i < MT * 16; i += NTHREADS) {
            const int j = i & 15;
            r2_lds[i] = (j < D_OUT) ? sigmoidf(u2_lds[i]) : 0.0f;
        }
        __syncthreads();

        // t-phase: t = r2 @ W2^T  (16 x 1024, K padded 10 -> 12), fused u1 update
        for (int nt = wave; nt < HID / 16; nt += 4) {
            const int n0 = nt * 16;
            const int h  = n0 + laneM;
            v8f tacc = {};
            #pragma unroll
            for (int k0 = 0; k0 < 12; k0 += 4) {
                const int k = k0 + hi * 2;
                v2f a; a.x = r2_lds[laneM * 16 + k]; a.y = r2_lds[laneM * 16 + k + 1];
                v2f b; b.x = w2_lds[h * W2P + k];    b.y = w2_lds[h * W2P + k + 1]; // W2^T[k][h]
                tacc = __builtin_amdgcn_wmma_f32_16x16x4_f32(
                    false, a, false, b, (short)0, tacc, false, false);
            }
            const float b1v = b1_lds[h];
            #pragma unroll
            for (int i = 0; i < 8; ++i) {          // C layout: M = i + 8*hi, N = h
                const int   mrow = i + hi * 8;
                const float u1v  = u1_lds[mrow * HID + h];
                const float xv   = xw1_lds[mrow * HID + h];
                const float r1   = sigmoidf(u1v);
                const float d1   = r1 * (1.0f - r1);
                const float g1   = u1v - d1 * (0.5f * xv + tacc[i] + b1v);
                u1_lds[mrow * HID + h] = u1v - 0.5f * g1;
                r1_lds[mrow * HID + h] = r1;
            }
        }
        __syncthreads();

        // s-phase: s = r1 @ W2  (16 x 16pad, K = 1024 split across 4 waves)
        {
            v8f sacc = {};
            const int kbase = wave * (HID / 4);
            for (int k0 = kbase; k0 < kbase + HID / 4; k0 += 4) {
                const int k = k0 + hi * 2;
                v2f a; a.x = r1_lds[laneM * HID + k]; a.y = r1_lds[laneM * HID + k + 1];
                v2f b; b.x = w2_lds[k * W2P + laneM]; b.y = w2_lds[(k + 1) * W2P + laneM];
                sacc = __builtin_amdgcn_wmma_f32_16x16x4_f32(
                    false, a, false, b, (short)0, sacc, false, false);
            }
            #pragma unroll
            for (int i = 0; i < 8; ++i)
                sred_lds[wave * 256 + (i + hi * 8) * 16 + laneM] = sacc[i];
        }
        __syncthreads();

        // u2 update (cross-wave reduce of the 4 partial s tiles)
        if (tid < MT * D_OUT) {
            const int m2 = tid / D_OUT, j2 = tid % D_OUT;
            float ssum = 0.0f;
            #pragma unroll
            for (int w = 0; w < 4; ++w)
                ssum += sred_lds[w * 256 + m2 * 16 + j2];
            const float u2v = u2_lds[m2 * 16 + j2];
            const float r2v = sigmoidf(u2v);
            const float d2  = r2v * (1.0f - r2v);
            const float g2  = u2v - d2 * (ssum + b2_lds[j2]);
            u2_lds[m2 * 16 + j2] = u2v - 0.5f * g2;
        }
        __syncthreads();
    }

    // ---------------- output: sigmoid(u2) --------------------------------
    if (tid < MT * D_OUT) {
        const int m2 = tid / D_OUT, j2 = tid % D_OUT;
        out[(size_t)(row0 + m2) * D_OUT + j2] = sigmoidf(u2_lds[m2 * 16 + j2]);
    }
}

extern "C" void kernel_launch(void* const* d_in, const int* in_sizes, int n_in,
                              void* d_out, int out_size, void* d_ws, size_t ws_size,
                              hipStream_t stream)
{
    const float* x   = (const float*)d_in[0];
    const float* u1  = (const float*)d_in[1];
    const float* u2  = (const float*)d_in[2];
    const float* W1  = (const float*)d_in[3];
    const float* W2  = (const float*)d_in[4];
    const float* b1  = (const float*)d_in[5];
    const float* b2  = (const float*)d_in[6];
    const int* steps = (const int*)d_in[7];
    float* out       = (float*)d_out;

    const int B = in_sizes[0] / D_IN;            // 16384
    const size_t lds_bytes =
        (size_t)(3 * MT * HID + HID * W2P + HID + 16 + 2 * MT * 16 + 4 * 256)
        * sizeof(float);

    eqprop_settle_kernel<<<dim3(B / MT), dim3(NTHREADS), lds_bytes, stream>>>(
        x, u1, u2, W1, W2, b1, b2, steps, out);
}